// GCN3_44418551775312
// MI455X (gfx1250) — compile-verified
//
#include <hip/hip_runtime.h>

typedef __attribute__((ext_vector_type(2))) float v2f;
typedef __attribute__((ext_vector_type(8))) float v8f;

#define N_NODES 50000
#define N_EDGES 800000
#define NFEAT   512
#define NHID    64
#define NCLASS  40
#define M_TILES (N_NODES / 16)   // 3125 exact

// ---------------------------------------------------------------------------
// GEMM: C[M x NT*16] = op(A)[M x K] * B[K x NB] + bias, via
// V_WMMA_F32_16X16X4_F32 (fp32-exact, matches reference precision).
//
// W (the B matrix) is staged once per workgroup into LDS (W1: 128 KB, fits
// the 320 KB WGP LDS with 2 workgroups resident), zero-padded to NT*16
// columns so the hot loop has no bounds selects. The WMMA inner loop then
// reads B via DS loads (co-executes with the matrix pipe) while A streams
// through the vector-memory path exactly once.
//
// Layouts per CDNA5 ISA 16x16x4 f32 WMMA: A 16x4 (lanes 0-15 hold K={0,1},
// lanes 16-31 hold K={2,3}); B 4x16 mirrored; C: VGPR j = rows j / j+8.
// ---------------------------------------------------------------------------
template <int NT, bool RELU_A, int NB, int K>
__global__ __launch_bounds__(256) void gemm_bias_wmma(
    const float* __restrict__ A, const float* __restrict__ B,
    const float* __restrict__ bias, float* __restrict__ C,
    int lda, int ldc)
{
    constexpr int LDBP = NT * 16;            // padded B row stride in LDS
    __shared__ float sB[K * LDBP];

    // Cooperative stage: B[K x NB] -> sB[K x LDBP], zero pad for n >= NB.
    for (int i = threadIdx.x; i < K * LDBP; i += 256) {
        const int kr = i / LDBP;             // LDBP constexpr -> cheap
        const int n  = i - kr * LDBP;
        float v = 0.0f;
        if (n < NB) v = B[kr * NB + n];      // folds away when NB == LDBP
        sB[i] = v;
    }
    __syncthreads();

    const int wave  = threadIdx.x >> 5;
    const int lane  = threadIdx.x & 31;
    const int tileM = blockIdx.x * 8 + wave;
    if (tileM >= M_TILES) return;            // wave-uniform; EXEC stays all-1s

    const int half = lane >> 4;              // 0 = lanes 0-15, 1 = lanes 16-31
    const int l15  = lane & 15;
    const int m0   = tileM * 16;

    v8f c[NT];
#pragma unroll
    for (int t = 0; t < NT; ++t) {
        const int n   = t * 16 + l15;
        const int ncl = (n < NB) ? n : 0;    // clamp; folds when NB == LDBP
        float bv      = bias[ncl];
        bv            = (n < NB) ? bv : 0.0f;
#pragma unroll
        for (int j = 0; j < 8; ++j) c[t][j] = bv;   // bias depends only on N
    }

    const float* arow = A + (long long)(m0 + l15) * lda;

    for (int kk = 0; kk < K; kk += 4) {
        const int kb = kk + half * 2;        // this half-wave's K pair
        v2f a = *(const v2f*)(arow + kb);
        if (RELU_A) {
            a.x = a.x > 0.0f ? a.x : 0.0f;
            a.y = a.y > 0.0f ? a.y : 0.0f;
        }
        const float* bbase = sB + kb * LDBP + l15;
#pragma unroll
        for (int t = 0; t < NT; ++t) {
            v2f b;
            b.x = bbase[t * 16];             // row kb   (ds load)
            b.y = bbase[LDBP + t * 16];      // row kb+1 (ds load)
            c[t] = __builtin_amdgcn_wmma_f32_16x16x4_f32(
                false, a, false, b, (short)0, c[t], false, false);
        }
    }

#pragma unroll
    for (int t = 0; t < NT; ++t) {
        const int n = t * 16 + l15;
#pragma unroll
        for (int j = 0; j < 8; ++j) {
            const int m = j + half * 8;      // C VGPR j -> row j / j+8
            C[(long long)(m0 + m) * ldc + n] = c[t][j];
        }
    }
}

// ---------------------------------------------------------------------------
// SpMM scatter: out[row[e], f] += support[col[e], f] * ew[e].
// Thread = (edge, 4-feature group): float4 gather + 4 atomic f32 adds.
// Destination is L2-resident (12.8 MB << 192 MB L2) so the atomics resolve
// in the L2 atomic units; the atomic count itself is irreducible.
// ---------------------------------------------------------------------------
template <int D, int LDS_, int LDO>
__global__ __launch_bounds__(256) void spmm_scatter(
    const float* __restrict__ sup, const int* __restrict__ row,
    const int* __restrict__ col, const float* __restrict__ ew,
    float* __restrict__ out)
{
    constexpr int D4 = D / 4;                // 64 -> 16, 40 -> 10
    const long long idx   = (long long)blockIdx.x * blockDim.x + threadIdx.x;
    const long long total = (long long)N_EDGES * D4;
    if (idx >= total) return;
    const int e  = (int)(idx / D4);          // compile-time divisor
    const int f  = (int)(idx - (long long)e * D4) * 4;
    const float w = ew[e];
    const float4 s = *(const float4*)(sup + (long long)col[e] * LDS_ + f);
    float* o = out + (long long)row[e] * LDO + f;
    atomicAdd(o + 0, s.x * w);
    atomicAdd(o + 1, s.y * w);
    atomicAdd(o + 2, s.z * w);
    atomicAdd(o + 3, s.w * w);
}

__global__ __launch_bounds__(256) void zero_f32(float* __restrict__ p, long long n)
{
    const long long i = (long long)blockIdx.x * blockDim.x + threadIdx.x;
    if (i < n) p[i] = 0.0f;
}

// Per-row log_softmax over NCLASS=40 (8 MB in + 8 MB out: trivially fast).
__global__ __launch_bounds__(256) void log_softmax_rows(
    const float* __restrict__ in, float* __restrict__ out)
{
    const int r = blockIdx.x * blockDim.x + threadIdx.x;
    if (r >= N_NODES) return;
    const float* x = in + (long long)r * NCLASS;
    float vals[NCLASS];
    float m = -3.402823466e38f;
#pragma unroll
    for (int i = 0; i < NCLASS; ++i) { vals[i] = x[i]; m = fmaxf(m, vals[i]); }
    float s = 0.0f;
#pragma unroll
    for (int i = 0; i < NCLASS; ++i) s += __expf(vals[i] - m);
    const float ls = m + __logf(s);
    float* o = out + (long long)r * NCLASS;
#pragma unroll
    for (int i = 0; i < NCLASS; ++i) o[i] = vals[i] - ls;
}

extern "C" void kernel_launch(void* const* d_in, const int* in_sizes, int n_in,
                              void* d_out, int out_size, void* d_ws, size_t ws_size,
                              hipStream_t stream)
{
    const float* x  = (const float*)d_in[0];
    const float* ew = (const float*)d_in[1];
    const float* W1 = (const float*)d_in[2];
    const float* b1 = (const float*)d_in[3];
    const float* W2 = (const float*)d_in[4];
    const float* b2 = (const float*)d_in[5];
    const float* W3 = (const float*)d_in[6];
    const float* b3 = (const float*)d_in[7];
    const int*  row = (const int*)d_in[8];
    const int*  col = (const int*)d_in[9];
    float* out = (float*)d_out;

    // Two ping-pong slots of 50000*64 floats (25.6 MB total workspace).
    float* slot0 = (float*)d_ws;
    float* slot1 = slot0 + (size_t)N_NODES * 64;

    const dim3 blk(256);
    const int  gemmBlocks = (M_TILES + 7) / 8;

    // ---- Layer 1: s1 = x @ W1 + b1 -> slot0 -------------------------------
    gemm_bias_wmma<4, false, NHID, NFEAT><<<gemmBlocks, blk, 0, stream>>>(
        x, W1, b1, slot0, NFEAT, NHID);
    {
        const long long n = (long long)N_NODES * 64;
        zero_f32<<<(unsigned)((n + 255) / 256), blk, 0, stream>>>(slot1, n);
        const long long t = (long long)N_EDGES * (64 / 4);
        spmm_scatter<64, 64, 64><<<(unsigned)((t + 255) / 256), blk, 0, stream>>>(
            slot0, row, col, ew, slot1);             // h1 -> slot1
    }

    // ---- Layer 2: s2 = relu(h1) @ W2 + b2 -> slot0 ------------------------
    gemm_bias_wmma<4, true, NHID, NHID><<<gemmBlocks, blk, 0, stream>>>(
        slot1, W2, b2, slot0, NHID, NHID);
    {
        const long long n = (long long)N_NODES * 64;
        zero_f32<<<(unsigned)((n + 255) / 256), blk, 0, stream>>>(slot1, n);
        const long long t = (long long)N_EDGES * (64 / 4);
        spmm_scatter<64, 64, 64><<<(unsigned)((t + 255) / 256), blk, 0, stream>>>(
            slot0, row, col, ew, slot1);             // h2 -> slot1
    }

    // ---- Layer 3: s3 = h2 @ W3 + b3 -> slot0 (48-wide padded, 40 valid) ---
    gemm_bias_wmma<3, false, NCLASS, NHID><<<gemmBlocks, blk, 0, stream>>>(
        slot1, W3, b3, slot0, NHID, 48);
    {
        const long long n = (long long)N_NODES * NCLASS;
        zero_f32<<<(unsigned)((n + 255) / 256), blk, 0, stream>>>(slot1, n);
        const long long t = (long long)N_EDGES * (NCLASS / 4);
        spmm_scatter<NCLASS, 48, NCLASS><<<(unsigned)((t + 255) / 256), blk, 0, stream>>>(
            slot0, row, col, ew, slot1);             // h3 -> slot1
    }

    // ---- log_softmax -> d_out ---------------------------------------------
    log_softmax_rows<<<(N_NODES + 255) / 256, blk, 0, stream>>>(slot1, out);
}